// GPCwSTU_11149735101051
// MI455X (gfx1250) — compile-verified
//
#include <hip/hip_runtime.h>
#include <hip/hip_bf16.h>
#include <math.h>

// Problem dims (match reference)
#define DD   256   // state dim
#define NN   64    // control dim
#define HH   8     // GPC history
#define TT   512   // horizon / steps
#define NF   20    // STU filters
#define ETA  1e-3f
#define DECAY 0.9f

typedef __attribute__((ext_vector_type(2))) float v2f;
typedef __attribute__((ext_vector_type(8))) float v8f;

// One persistent workgroup runs the whole sequential horizon.
// State kept in LDS; M is represented exactly as M[h] = alpha[h]*M0[h] + B[h,:] (x) w[h,:].
__launch_bounds__(1024, 1)
__global__ void gpc_stu_kernel(const float* __restrict__ Q,     // [D,D]
                               const float* __restrict__ R,     // [N,N]
                               const float* __restrict__ K,     // [N,D]
                               const float* __restrict__ M0,    // [H,N,D]
                               const float* __restrict__ Mstu,  // [NF,D,N]
                               const float* __restrict__ x0,    // [D]
                               const float* __restrict__ phi,   // [T,NF]
                               const float* __restrict__ w,     // [H,D]
                               float* __restrict__ losses,      // [T] out
                               float* __restrict__ uhist)       // [N,T] scratch (circular by step)
{
    __shared__ float s_phi[TT * NF];     // 40960 B, B-matrix cache for WMMA (row-major [j][k])
    __shared__ float s_proj[NN * 32];    // 8192 B, padded [64,32]
    __shared__ float s_x[DD];
    __shared__ float s_qx[DD];
    __shared__ float s_u[NN];
    __shared__ float s_kx[NN];
    __shared__ float s_ru[NN];
    __shared__ float s_mterm[NN];
    __shared__ float s_B[HH * NN];       // rank-1 coefficient state
    __shared__ float s_Bt[HH * NN];      // post-SGD, pre-projection
    __shared__ float s_m0w[HH * NN];     // M0[h] @ w[h]
    __shared__ float s_alpha[HH];
    __shared__ float s_ww[HH];           // ||w[h]||^2
    __shared__ float s_q0[HH];           // ||M0[h]||_F^2
    __shared__ float s_scale[HH];
    __shared__ float s_red[1024];        // shared reduction / partial buffer

    const int tid  = threadIdx.x;
    const int lane = tid & 31;
    const int wave = tid >> 5;

    // ----- one-time init -----
    for (int i = tid; i < TT * NF; i += 1024) s_phi[i] = phi[i];
    for (int i = tid; i < NN * TT; i += 1024) uhist[i] = 0.0f;   // future slots must be zero
    for (int i = tid; i < DD; i += 1024)     s_x[i] = x0[i];
    if (tid < HH * NN) s_B[tid] = 0.0f;
    if (tid < HH)      s_alpha[tid] = 1.0f;

    if (tid < HH * NN) {                  // tid = h*64 + n
        const float* m0 = M0 + tid * DD;
        const int h = tid >> 6;
        const float* whp = w + h * DD;
        float mw = 0.f, m2 = 0.f;
        for (int d = 0; d < DD; ++d) {
            float m = m0[d];
            mw = fmaf(m, whp[d], mw);
            m2 = fmaf(m, m, m2);
        }
        s_m0w[tid] = mw;
        s_red[tid] = m2;                  // partial for q0
    }
    if (tid < HH) {
        const float* whp = w + tid * DD;
        float acc = 0.f;
        for (int d = 0; d < DD; ++d) acc = fmaf(whp[d], whp[d], acc);
        s_ww[tid] = acc;
    }
    __syncthreads();
    if (tid < HH) {
        float q = 0.f;
        for (int n = 0; n < NN; ++n) q += s_red[tid * NN + n];
        s_q0[tid] = q;
    }
    __syncthreads();

    // ----- sequential horizon -----
    for (int t = 0; t < TT; ++t) {
        // Phase A: Qx, Kx, M-contribution to u
        if (tid < DD) {
            const float* q = Q + tid * DD;
            __builtin_prefetch(q, 0, 3);
            float acc = 0.f;
            for (int e = 0; e < DD; ++e) acc = fmaf(q[e], s_x[e], acc);
            s_qx[tid] = acc;
        } else if (tid < DD + NN) {
            const int n = tid - DD;
            const float* kk = K + n * DD;
            float acc = 0.f;
            for (int e = 0; e < DD; ++e) acc = fmaf(kk[e], s_x[e], acc);
            s_kx[n] = acc;
        } else if (tid < DD + 2 * NN) {
            const int n = tid - DD - NN;
            float acc = 0.f;
            for (int h = 0; h < HH; ++h)
                acc += s_alpha[h] * s_m0w[h * NN + n] + s_B[h * NN + n] * s_ww[h];
            s_mterm[n] = acc;
        }
        __syncthreads();

        // Phase B: u = -Kx + mterm ; push into circular history slot t
        if (tid < NN) {
            float uu = s_mterm[tid] - s_kx[tid];
            s_u[tid] = uu;
            uhist[tid * TT + t] = uu;
        }
        __threadfence_block();
        __syncthreads();

        // Phase C: Ru and loss partials
        if (tid < NN) {
            const float* rr = R + tid * NN;
            float acc = 0.f;
            for (int m = 0; m < NN; ++m) acc = fmaf(rr[m], s_u[m], acc);
            s_ru[tid] = acc;
        }
        if (tid < DD) s_red[tid] = s_x[tid] * s_qx[tid];
        __syncthreads();
        if (tid == 0) {
            float c = 0.f;
            for (int i = 0; i < DD; ++i) c += s_red[i];
            for (int n = 0; n < NN; ++n) c += s_u[n] * s_ru[n];
            losses[t] = c;
        }
        __syncthreads();   // protect s_red before reuse

        // Phase D: SGD + Frobenius projection in (alpha, B) factored form
        if (tid < HH * NN) {               // tid = h*64 + n
            float bt = s_B[tid] - (ETA * 2.0f) * s_ru[tid & 63];
            s_Bt[tid] = bt;
            s_red[tid]       = bt * s_m0w[tid];   // <M0[h], bt (x) w[h]> partial
            s_red[512 + tid] = bt * bt;           // ||bt||^2 partial
        }
        __syncthreads();
        if (tid < HH) {
            float p1 = 0.f, p2 = 0.f;
            for (int n = 0; n < NN; ++n) {
                p1 += s_red[tid * NN + n];
                p2 += s_red[512 + tid * NN + n];
            }
            float a = s_alpha[tid];
            float nrm2 = a * a * s_q0[tid] + 2.0f * a * p1 + p2 * s_ww[tid];
            float nrm = sqrtf(nrm2);
            float limit = powf(DECAY, (float)t);
            float sc = (nrm > limit) ? (limit / nrm) : 1.0f;
            s_scale[tid] = sc;
            s_alpha[tid] = a * sc;
        }
        __syncthreads();
        if (tid < HH * NN) s_B[tid] = s_Bt[tid] * s_scale[tid >> 6];

        // Phase E: proj[n,k] = sum_{j<=t} u_{t-j}[n] * phi[j,k]  via V_WMMA_F32_16X16X4_F32
        // 8 waves, one 16x16 output tile each (M: 4 tiles of n, N: 2 tiles of k padded to 32).
        // Unconditional loads: B column index clamped + scaled by {0,1}; A index wraps onto
        // guaranteed-zero future slots of the circular history.
        if (wave < 8) {
            const int mt   = wave >> 1;        // n-tile 0..3
            const int nt   = wave & 1;         // k-tile 0..1
            const int half = lane >> 4;        // K-pair select within 16x16x4 step
            const int lid  = lane & 15;
            const int n    = mt * 16 + lid;    // A row (control index)
            const int kcol = nt * 16 + lid;    // B col (filter index)
            const int kc   = (kcol < NF) ? kcol : (NF - 1);
            const float bsel = (kcol < NF) ? 1.0f : 0.0f;
            const float* uh = uhist + n * TT;
            const float* ph = s_phi + kc;      // column base, row stride NF
            v8f c = {};
            const int jmax = (t + 4) & ~3;     // only j <= t contributes
            for (int j0 = 0; j0 < jmax; j0 += 4) {
                const int ja = j0 + 2 * half;  // this lane-half holds K = ja, ja+1
                v2f a, b;
                a.x = uh[(t - ja)     & (TT - 1)];   // u_{t-j}[n]; future slots are zero
                a.y = uh[(t - ja - 1) & (TT - 1)];
                b.x = ph[ja * NF]       * bsel;
                b.y = ph[(ja + 1) * NF] * bsel;
                c = __builtin_amdgcn_wmma_f32_16x16x4_f32(
                        /*neg_a=*/false, a, /*neg_b=*/false, b,
                        /*c_mod=*/(short)0, c, /*reuse_a=*/false, /*reuse_b=*/false);
            }
            #pragma unroll
            for (int v = 0; v < 8; ++v) {
                const int m = mt * 16 + v + half * 8;    // C layout: VGPR v -> M=v (+8 hi half)
                s_proj[m * 32 + kcol] = c[v];
            }
        }
        __syncthreads();

        // Phase F: x_next[d] = sum_{k<20} sum_{n<64} Mstu[k,d,n] * proj[n,k]
        {
            const int d    = tid & 255;
            const int part = tid >> 8;           // 4 parts x 5 filters
            float acc = 0.f;
            for (int k = part * 5; k < part * 5 + 5; ++k) {
                const float* ms = Mstu + (k * DD + d) * NN;
                __builtin_prefetch(ms, 0, 3);
                for (int n = 0; n < NN; ++n)
                    acc = fmaf(ms[n], s_proj[n * 32 + k], acc);
            }
            s_red[part * DD + d] = acc;
        }
        __syncthreads();
        if (tid < DD)
            s_x[tid] = s_red[tid] + s_red[DD + tid] + s_red[2 * DD + tid] + s_red[3 * DD + tid];
        __syncthreads();
    }
}

extern "C" void kernel_launch(void* const* d_in, const int* in_sizes, int n_in,
                              void* d_out, int out_size, void* d_ws, size_t ws_size,
                              hipStream_t stream) {
    const float* Q    = (const float*)d_in[0];   // [256,256]
    const float* R    = (const float*)d_in[1];   // [64,64]
    const float* K    = (const float*)d_in[2];   // [64,256]
    const float* M0   = (const float*)d_in[3];   // [8,64,256]
    const float* Mstu = (const float*)d_in[4];   // [20,256,64]
    const float* x0   = (const float*)d_in[5];   // [256,1]
    const float* phi  = (const float*)d_in[6];   // [512,20]
    const float* wh   = (const float*)d_in[7];   // [8,256]
    float* losses = (float*)d_out;               // [512]
    float* uhist  = (float*)d_ws;                // [64,512] scratch (128 KB)

    gpc_stu_kernel<<<1, 1024, 0, stream>>>(Q, R, K, M0, Mstu, x0, phi, wh, losses, uhist);
}